// FixedTripletLoss_13460427505719
// MI455X (gfx1250) — compile-verified
//
#include <hip/hip_runtime.h>
#include <hip/hip_bf16.h>
#include <math.h>

typedef float v2f __attribute__((ext_vector_type(2)));
typedef float v8f __attribute__((ext_vector_type(8)));
typedef int   v4i __attribute__((ext_vector_type(4)));

typedef __attribute__((address_space(1))) v4i gv4i;  // global int4
typedef __attribute__((address_space(3))) v4i lv4i;  // LDS int4

#define N_ROWS 8192
#define DIM    256
#define MARGIN 0.3f

// Direct global -> LDS async copy (16B per lane), tracked by ASYNCcnt.
__device__ __forceinline__ void async_copy_b128(const float* g, float* l) {
  __builtin_amdgcn_global_load_async_to_lds_b128(
      (gv4i*)g, (lv4i*)l, /*offset=*/0, /*cpol=*/0);
}
__device__ __forceinline__ void wait_async0() {
  asm volatile("s_wait_asynccnt 0x0" ::: "memory");
}

// ---------------------------------------------------------------------------
// Kernel 1: L2-normalize rows, store normalized matrix Xn and the squared
// norm of each *normalized* row (matches reference, which recomputes sq_norms
// from the normalized x).  One wave per row; 8 rows per 256-thread block.
// ---------------------------------------------------------------------------
__global__ __launch_bounds__(256) void normalize_kernel(
    const float* __restrict__ emb, float* __restrict__ xn,
    float* __restrict__ sqn) {
  const int lane = threadIdx.x & 31;
  const int wave = threadIdx.x >> 5;
  const int row  = blockIdx.x * 8 + wave;
  const float* src = emb + (size_t)row * DIM;
  float*       dst = xn  + (size_t)row * DIM;

  float v[8];
  float s = 0.f;
#pragma unroll
  for (int i = 0; i < 8; ++i) {
    v[i] = src[lane + 32 * i];
    s += v[i] * v[i];
  }
#pragma unroll
  for (int off = 16; off; off >>= 1) s += __shfl_xor(s, off, 32);

  const float inv = 1.0f / fmaxf(sqrtf(s), 1e-12f);
  float s2 = 0.f;
#pragma unroll
  for (int i = 0; i < 8; ++i) {
    float t = v[i] * inv;
    dst[lane + 32 * i] = t;
    s2 += t * t;
  }
#pragma unroll
  for (int off = 16; off; off >>= 1) s2 += __shfl_xor(s2, off, 32);
  if (lane == 0) sqn[row] = s2;
}

// ---------------------------------------------------------------------------
// Kernel 2: fused Gram-matrix (fp32 WMMA) + batch-hard mining.
// 256 threads = 8 waves as 2(M) x 4(N) 16x16 WMMA tiles -> 32 anchors x 64
// cols per iteration.  Full-K A tile staged in LDS ONCE; full-K B tile staged
// per iteration via GLOBAL_LOAD_ASYNC_TO_LDS_B128 (no VGPR round-trip).
// Dual accumulators break the WMMA RAW chain.  Stride 260: 16B-aligned rows
// + conflict-free ds_load_b64 fragment reads.
// ---------------------------------------------------------------------------
#define ASTR 260

__global__ __launch_bounds__(256) void triplet_mine_kernel(
    const float* __restrict__ xn, const float* __restrict__ sqn,
    const int* __restrict__ labels, float* __restrict__ pa_out,
    float* __restrict__ valid_out) {
  __shared__ __align__(16) float As[32 * ASTR];  // 33,280 B
  __shared__ __align__(16) float Bs[64 * ASTR];  // 66,560 B
  __shared__ float redAp[32][4];
  __shared__ float redAn[32][4];

  const int tid   = threadIdx.x;
  const int lane  = tid & 31;
  const int wave  = tid >> 5;
  const int waveM = wave >> 2;   // 0..1
  const int waveN = wave & 3;    // 0..3
  const int half  = lane >> 4;   // 0..1
  const int l16   = lane & 15;

  const int anchorBase = blockIdx.x * 32;

  // C-tile layout: VGPR v holds row M = waveM*16 + v + 8*half, N = l16.
  int   rowIdx[8];
  int   rowLab[8];
  float rowSq[8];
#pragma unroll
  for (int v = 0; v < 8; ++v) {
    int r = anchorBase + waveM * 16 + v + 8 * half;
    rowIdx[v] = r;
    rowLab[v] = labels[r];
    rowSq[v]  = sqn[r];
  }

  float runAp[8], runAn[8];
#pragma unroll
  for (int v = 0; v < 8; ++v) {
    runAp[v] = -INFINITY;
    runAn[v] =  INFINITY;
  }

  // Stage the block-invariant A tile (32 rows x 256 K) once, async.
#pragma unroll
  for (int i = 0; i < 8; ++i) {
    int idx = tid + 256 * i;   // 0..2047
    int r = idx >> 6, c4 = idx & 63;
    async_copy_b128(&xn[(size_t)(anchorBase + r) * DIM + c4 * 4],
                    &As[r * ASTR + c4 * 4]);
  }

  const int aRowBase = (waveM * 16 + l16) * ASTR + 2 * half;
  const int bRowBase = (waveN * 16 + l16) * ASTR + 2 * half;

  for (int j = 0; j < N_ROWS; j += 64) {
    __syncthreads();  // previous B fully consumed
    // Stage B tile (64 rows x 256 K) async: 16 x b128 per thread.
#pragma unroll
    for (int i = 0; i < 16; ++i) {
      int idx = tid + 256 * i;
      int r = idx >> 6, c4 = idx & 63;
      async_copy_b128(&xn[(size_t)(j + r) * DIM + c4 * 4],
                      &Bs[r * ASTR + c4 * 4]);
    }
    wait_async0();    // this wave's async copies (A's too, on iter 0) done
    __syncthreads();  // all waves' copies visible

    // 64 K-steps of 4; two independent accumulator chains.
    v8f acc0 = {}, acc1 = {};
#pragma unroll
    for (int t = 0; t < 64; t += 2) {
      v2f a0 = *(const v2f*)&As[aRowBase + 4 * t];
      v2f b0 = *(const v2f*)&Bs[bRowBase + 4 * t];
      v2f a1 = *(const v2f*)&As[aRowBase + 4 * t + 4];
      v2f b1 = *(const v2f*)&Bs[bRowBase + 4 * t + 4];
      acc0 = __builtin_amdgcn_wmma_f32_16x16x4_f32(
          false, a0, false, b0, (short)0, acc0, false, false);
      acc1 = __builtin_amdgcn_wmma_f32_16x16x4_f32(
          false, a1, false, b1, (short)0, acc1, false, false);
    }

    // Epilogue for this 16x16 tile: this lane's column.
    const int c    = j + waveN * 16 + l16;
    const int clab = labels[c];
    const float csq = sqn[c];
#pragma unroll
    for (int v = 0; v < 8; ++v) {
      float dot = acc0[v] + acc1[v];
      float sq = rowSq[v] + csq - 2.0f * dot;
      sq = fmaxf(sq, 0.0f);
      float d = (sq > 0.0f) ? sqrtf(sq) : 0.0f;
      if (rowLab[v] == clab) {
        if (rowIdx[v] != c) runAp[v] = fmaxf(runAp[v], d);
      } else {
        runAn[v] = fminf(runAn[v], d);
      }
    }
  }

  // Butterfly reduce over 16 lanes per half (halves hold disjoint M rows).
#pragma unroll
  for (int v = 0; v < 8; ++v) {
#pragma unroll
    for (int off = 8; off; off >>= 1) {
      runAp[v] = fmaxf(runAp[v], __shfl_xor(runAp[v], off, 32));
      runAn[v] = fminf(runAn[v], __shfl_xor(runAn[v], off, 32));
    }
  }
  if (l16 == 0) {
#pragma unroll
    for (int v = 0; v < 8; ++v) {
      int lr = waveM * 16 + v + 8 * half;
      redAp[lr][waveN] = runAp[v];
      redAn[lr][waveN] = runAn[v];
    }
  }
  __syncthreads();

  if (tid < 32) {
    float ap = redAp[tid][0], an = redAn[tid][0];
#pragma unroll
    for (int q = 1; q < 4; ++q) {
      ap = fmaxf(ap, redAp[tid][q]);
      an = fminf(an, redAn[tid][q]);
    }
    const bool valid = (ap > -INFINITY) && (an < INFINITY);
    const float loss = valid ? fmaxf(ap - an + MARGIN, 0.0f) : 0.0f;
    pa_out[anchorBase + tid]    = loss;
    valid_out[anchorBase + tid] = valid ? 1.0f : 0.0f;
  }
}

// ---------------------------------------------------------------------------
// Kernel 3: deterministic single-block reduction (fixed order, bit-identical
// across graph replays).  out[0] = loss, out[1] = num_valid.
// ---------------------------------------------------------------------------
__global__ __launch_bounds__(256) void finalize_kernel(
    const float* __restrict__ pa, const float* __restrict__ vf,
    float* __restrict__ out) {
  __shared__ float sl[256];
  __shared__ float sv[256];
  const int tid = threadIdx.x;
  float a = 0.f, b = 0.f;
#pragma unroll
  for (int i = 0; i < N_ROWS / 256; ++i) {
    a += pa[tid + 256 * i];
    b += vf[tid + 256 * i];
  }
  sl[tid] = a;
  sv[tid] = b;
  __syncthreads();
  for (int off = 128; off; off >>= 1) {
    if (tid < off) {
      sl[tid] += sl[tid + off];
      sv[tid] += sv[tid + off];
    }
    __syncthreads();
  }
  if (tid == 0) {
    float nv = sv[0];
    out[0] = (nv > 0.0f) ? (sl[0] / fmaxf(nv, 1.0f)) : 0.0f;
    out[1] = nv;
  }
}

// ---------------------------------------------------------------------------
extern "C" void kernel_launch(void* const* d_in, const int* in_sizes, int n_in,
                              void* d_out, int out_size, void* d_ws, size_t ws_size,
                              hipStream_t stream) {
  const float* emb    = (const float*)d_in[0];
  const int*   labels = (const int*)d_in[1];
  float* out = (float*)d_out;

  // Workspace layout (16B-aligned slices).
  char* ws = (char*)d_ws;
  float* xn  = (float*)ws;                                   // 8192*256 f32
  float* sqn = (float*)(ws + (size_t)N_ROWS * DIM * 4);      // 8192 f32
  float* pa  = sqn + N_ROWS;                                 // 8192 f32
  float* vf  = pa + N_ROWS;                                  // 8192 f32

  normalize_kernel<<<N_ROWS / 8, 256, 0, stream>>>(emb, xn, sqn);
  triplet_mine_kernel<<<N_ROWS / 32, 256, 0, stream>>>(xn, sqn, labels, pa, vf);
  finalize_kernel<<<1, 256, 0, stream>>>(pa, vf, out);
}